// MultiGaussianDiffFusion_7241314861651
// MI455X (gfx1250) — compile-verified
//
#include <hip/hip_runtime.h>
#include <math.h>

// ---------------- tile geometry ----------------
#define HW       512
#define TX       64          // output tile width
#define TY       32          // output tile height
#define IN_H     (TY + 6)    // 38 rows incl. +/-3 vertical halo
#define IN_W     72          // TX + 8 cols (4-col halo each side -> b128 aligned)
#define NSLOT    (IN_H * (IN_W / 4))   // float4 transfer slots = 684
#define NTHREADS 256

struct Taps { float g3[3]; float g5[5]; float g7[7]; };

// clang vector types: v4i matches the async-LDS builtin's parameter type
typedef int   v4i __attribute__((vector_size(16)));
typedef float v4f __attribute__((ext_vector_type(4)));

// ---------------- CDNA5 async global->LDS path ----------------
#if defined(__has_builtin)
# if __has_builtin(__builtin_amdgcn_global_load_async_to_lds_b128)
#  define HAVE_ASYNC_LDS 1
# endif
#endif
#ifndef HAVE_ASYNC_LDS
# define HAVE_ASYNC_LDS 0
#endif

__device__ __forceinline__ void wait_asynccnt0() {
#if defined(__has_builtin)
# if __has_builtin(__builtin_amdgcn_s_wait_asynccnt)
  __builtin_amdgcn_s_wait_asynccnt(0);
# else
  asm volatile("s_wait_asynccnt 0" ::: "memory");
# endif
#else
  asm volatile("s_wait_asynccnt 0" ::: "memory");
#endif
}

__device__ __forceinline__ v4f ld4s(const float* p) { return *(const v4f*)p; }

// =====================================================================
// Fused separable multi-Gaussian band-pass:
//   out = (w1+w2)*s + (w3-w2)*G3(s) + (w4-w3)*G5(s) + (-w4)*G7(s)
// One LDS-staged tile (async global->LDS), one horizontal pass
// producing h3/h5/h7, one vertical pass with fused combine, NT stores.
// =====================================================================
__global__ __launch_bounds__(NTHREADS)
void mgdf_kernel(const float* __restrict__ src,
                 const float* __restrict__ w1p, const float* __restrict__ w2p,
                 const float* __restrict__ w3p, const float* __restrict__ w4p,
                 float* __restrict__ dst, Taps tp)
{
  __shared__ float s_in[IN_H][IN_W];   // input tile + halo (16B aligned rows)
  __shared__ float s_h3[IN_H][TX];     // horizontal blur results
  __shared__ float s_h5[IN_H][TX];
  __shared__ float s_h7[IN_H][TX];

  const int tid = threadIdx.x;
  const int x0  = blockIdx.x * TX;
  const int y0  = blockIdx.y * TY;
  const float* sp = src + (size_t)blockIdx.z * (HW * (size_t)HW);
  float*       dp = dst + (size_t)blockIdx.z * (HW * (size_t)HW);

  // uniform scalar weights -> s_load_b32 from constant cache (no lane games)
  const float W1 = *w1p, W2 = *w2p, W3 = *w3p, W4 = *w4p;

  // ---------------- stage input tile (zero-padded) ----------------
#if HAVE_ASYNC_LDS
  const bool boundary = (x0 == 0) | (x0 + TX == HW) | (y0 == 0) | (y0 + TY == HW);
  if (boundary) {
    const v4f z = (v4f)0.f;
    for (int p = tid; p < NSLOT; p += NTHREADS)
      ((v4f*)&s_in[0][0])[p] = z;
    __syncthreads();   // uniform per block
  }
  for (int p = tid; p < NSLOT; p += NTHREADS) {
    const int row = p / (IN_W / 4);
    const int gx  = x0 - 4 + (p % (IN_W / 4)) * 4;   // slots fully in or fully out
    const int gy  = y0 - 3 + row;
    if ((unsigned)gy < (unsigned)HW && gx >= 0 && gx < HW) {
      __builtin_amdgcn_global_load_async_to_lds_b128(
          (__attribute__((address_space(1))) v4i*)(sp + (size_t)gy * HW + gx),
          (__attribute__((address_space(3))) v4i*)(&((float*)s_in)[p * 4]),
          0, 0);
    }
  }
  wait_asynccnt0();
  __syncthreads();
#else
  for (int p = tid; p < NSLOT; p += NTHREADS) {
    const int row = p / (IN_W / 4);
    const int gx  = x0 - 4 + (p % (IN_W / 4)) * 4;
    const int gy  = y0 - 3 + row;
    v4f v = (v4f)0.f;
    if ((unsigned)gy < (unsigned)HW && gx >= 0 && gx < HW)
      v = ld4s(sp + (size_t)gy * HW + gx);
    ((v4f*)&s_in[0][0])[p] = v;
  }
  __syncthreads();
#endif

  // ---------------- horizontal pass: 15 taps -> h3/h5/h7 ----------------
  const float g30 = tp.g3[0], g31 = tp.g3[1], g32 = tp.g3[2];
  const float g50 = tp.g5[0], g51 = tp.g5[1], g52 = tp.g5[2], g53 = tp.g5[3], g54 = tp.g5[4];
  const float g70 = tp.g7[0], g71 = tp.g7[1], g72 = tp.g7[2], g73 = tp.g7[3],
              g74 = tp.g7[4], g75 = tp.g7[5], g76 = tp.g7[6];

  for (int p = tid; p < IN_H * TX; p += NTHREADS) {
    const int row = p >> 6;            // / TX
    const int c   = p & (TX - 1);      // output col; center at s_in col c+4
    const float* r = &s_in[row][c + 1];   // r[0..6] = taps -3..+3
    const float a0 = r[0], a1 = r[1], a2 = r[2], a3 = r[3], a4 = r[4], a5 = r[5], a6 = r[6];
    s_h3[row][c] = fmaf(g30, a2, fmaf(g31, a3, g32 * a4));
    s_h5[row][c] = fmaf(g50, a1, fmaf(g51, a2, fmaf(g52, a3, fmaf(g53, a4, g54 * a5))));
    s_h7[row][c] = fmaf(g70, a0, fmaf(g71, a1, fmaf(g72, a2,
                   fmaf(g73, a3, fmaf(g74, a4, fmaf(g75, a5, g76 * a6))))));
  }
  __syncthreads();

  // ---------------- vertical pass + fused combine ----------------
  const float ca = W1 + W2, cb = W3 - W2, cc = W4 - W3, cd = -W4;

#pragma unroll
  for (int it = 0; it < (TY * (TX / 4)) / NTHREADS; ++it) {   // exactly 2
    const int p  = tid + it * NTHREADS;
    const int r  = p >> 4;              // / (TX/4)
    const int c4 = (p & 15) * 4;

    v4f v3 = g30 * ld4s(&s_h3[r + 2][c4]);
    v3 += g31 * ld4s(&s_h3[r + 3][c4]);
    v3 += g32 * ld4s(&s_h3[r + 4][c4]);

    v4f v5 = g50 * ld4s(&s_h5[r + 1][c4]);
    v5 += g51 * ld4s(&s_h5[r + 2][c4]);
    v5 += g52 * ld4s(&s_h5[r + 3][c4]);
    v5 += g53 * ld4s(&s_h5[r + 4][c4]);
    v5 += g54 * ld4s(&s_h5[r + 5][c4]);

    v4f v7 = g70 * ld4s(&s_h7[r + 0][c4]);
    v7 += g71 * ld4s(&s_h7[r + 1][c4]);
    v7 += g72 * ld4s(&s_h7[r + 2][c4]);
    v7 += g73 * ld4s(&s_h7[r + 3][c4]);
    v7 += g74 * ld4s(&s_h7[r + 4][c4]);
    v7 += g75 * ld4s(&s_h7[r + 5][c4]);
    v7 += g76 * ld4s(&s_h7[r + 6][c4]);

    v4f o = ca * ld4s(&s_in[r + 3][c4 + 4]);
    o += cb * v3;
    o += cc * v5;
    o += cd * v7;

    // write-once output: non-temporal b128 store keeps L2 for the reused input
    __builtin_nontemporal_store(o, (v4f*)(dp + (size_t)(y0 + r) * HW + (x0 + c4)));
  }
}

// ---------------- host side ----------------
static void gauss1d(int k, float* o) {
  // matches reference: sigma = 0.3*((k-1)*0.5 - 1) + 0.8; normalized 1-D kernel
  const double sigma = 0.3 * ((k - 1) * 0.5 - 1.0) + 0.8;
  double sum = 0.0;
  for (int i = 0; i < k; ++i) {
    const double x = (double)(i - k / 2);
    const double v = exp(-(x * x) / (2.0 * sigma * sigma));
    o[i] = (float)v; sum += v;
  }
  for (int i = 0; i < k; ++i) o[i] = (float)((double)o[i] / sum);
}

extern "C" void kernel_launch(void* const* d_in, const int* in_sizes, int n_in,
                              void* d_out, int out_size, void* d_ws, size_t ws_size,
                              hipStream_t stream) {
  (void)n_in; (void)out_size; (void)d_ws; (void)ws_size;
  const float* s  = (const float*)d_in[0];
  const float* w1 = (const float*)d_in[1];
  const float* w2 = (const float*)d_in[2];
  const float* w3 = (const float*)d_in[3];
  const float* w4 = (const float*)d_in[4];
  float* out = (float*)d_out;

  const int planes = in_sizes[0] / (HW * HW);   // B*C = 192

  Taps tp;
  gauss1d(3, tp.g3);
  gauss1d(5, tp.g5);
  gauss1d(7, tp.g7);

  dim3 grid(HW / TX, HW / TY, planes);   // 8 x 16 x 192
  dim3 block(NTHREADS);
  mgdf_kernel<<<grid, block, 0, stream>>>(s, w1, w2, w3, w4, out, tp);
}